// ROPE_55362128445910
// MI455X (gfx1250) — compile-verified
//
#include <hip/hip_runtime.h>

typedef __attribute__((ext_vector_type(2))) float v2f;
typedef __attribute__((ext_vector_type(8))) float v8f;

#define NBLK 2048
#define NTHR 256

// ---------------------------------------------------------------------------
// Sum of the 32 lane values of a wave, broadcast to all lanes.
// Uses V_WMMA_F32_16X16X4_F32: A holds the lane partials (one real K-slot per
// lane, rest zero), B = all-ones, C = 0.  Then D[m][n] = rowsum(A[m]) for every
// n.  Per the CDNA5 C/D layout (lanes 0-15 hold M=0..7 in VGPRs 0..7, lanes
// 16-31 hold M=8..15), summing one lane's 8 D registers yields the half-wave
// row-sum total; a single xor-16 shuffle adds the two halves.  This is
// layout-robust: it only relies on the documented C/D mapping, since B==1
// makes D independent of the A/B intra-layout details.
// Requires EXEC == all ones (callers guarantee full waves, no divergence).
// ---------------------------------------------------------------------------
__device__ __forceinline__ float wave_sum32(float v) {
#if __has_builtin(__builtin_amdgcn_wmma_f32_16x16x4_f32)
  v2f a; a[0] = v;    a[1] = 0.0f;
  v2f b; b[0] = 1.0f; b[1] = 1.0f;
  v8f c = {};
  v8f d = __builtin_amdgcn_wmma_f32_16x16x4_f32(
      /*neg_a=*/false, a, /*neg_b=*/false, b,
      /*c_mod=*/(short)0, c, /*reuse_a=*/false, /*reuse_b=*/false);
  float t = ((d[0] + d[1]) + (d[2] + d[3])) + ((d[4] + d[5]) + (d[6] + d[7]));
  t += __shfl_xor(t, 16, 32);
  return t;
#else
  for (int off = 16; off > 0; off >>= 1) v += __shfl_xor(v, off, 32);
  return v;
#endif
}

// 5-iteration CORDIC angle + 1.57*sin, exactly mirroring the reference
// (fp32 throughout; constants are float-rounded atan(2^-k)).
__device__ __forceinline__ float cordic_est(float th) {
  float cr;
  cr  = (0.0f > th) ? -0.7853981633974483f  : 0.7853981633974483f;
  cr += (cr   > th) ? -0.46364760900080615f : 0.46364760900080615f;
  cr += (cr   > th) ? -0.24497866312686414f : 0.24497866312686414f;
  cr += (cr   > th) ? -0.12435499454676144f : 0.12435499454676144f;
  cr += (cr   > th) ? -0.06241880999595735f : 0.06241880999595735f;
  // |cr| <= ~1.74 rad -> hardware v_sin_f32 is plenty accurate here.
  return 1.57f * __sinf(cr);
}

// ---------------------------------------------------------------------------
// Kernel 1: streaming partial reduction.  Each thread grid-strides over
// float4 chunks (two (v0,v1) pairs per load pair), accumulates
//   e0*(v0 - v1) + e1*(v0 + v1)
// in fp32, then the block reduces (WMMA wave-sum + tiny LDS step) and writes
// one partial per block into d_ws.
// ---------------------------------------------------------------------------
__global__ __launch_bounds__(NTHR) void rope_partial_kernel(
    const float* __restrict__ vec, const float* __restrict__ thetas,
    const int* __restrict__ mp, float* __restrict__ part, int n) {
  const float m = (float)mp[0];
  const int tid = blockIdx.x * NTHR + threadIdx.x;
  const int nth = gridDim.x * NTHR;
  const int n4  = n >> 2;

  const float4* __restrict__ v4 = (const float4*)vec;
  const float4* __restrict__ t4 = (const float4*)thetas;

  float acc = 0.0f;
  for (int i = tid; i < n4; i += nth) {
    // Pull the next stride's lines toward the caches (speculative; OOB
    // prefetches are silently dropped).  -> global_prefetch_b8
    __builtin_prefetch(v4 + i + nth, 0, 0);
    __builtin_prefetch(t4 + i + nth, 0, 0);

    float4 v = v4[i];
    float4 t = t4[i];
    float e0 = cordic_est(m * t.x);
    float e1 = cordic_est(m * t.y);
    float e2 = cordic_est(m * t.z);
    float e3 = cordic_est(m * t.w);
    acc = fmaf(e0, v.x - v.y, acc);
    acc = fmaf(e1, v.x + v.y, acc);
    acc = fmaf(e2, v.z - v.w, acc);
    acc = fmaf(e3, v.z + v.w, acc);
  }
  // Tail pairs (n is always even; E is a multiple of 4 in practice).
  if (tid == 0) {
    for (int i = n4 << 2; i + 1 < n; i += 2) {
      float e0 = cordic_est(m * thetas[i]);
      float e1 = cordic_est(m * thetas[i + 1]);
      acc = fmaf(e0, vec[i] - vec[i + 1], acc);
      acc = fmaf(e1, vec[i] + vec[i + 1], acc);
    }
  }

  // All 256 threads reach here with full EXEC -> WMMA wave reduction is legal.
  float wsum = wave_sum32(acc);

  __shared__ float lds[NTHR / 32];
  const int lane = threadIdx.x & 31;
  const int wid  = threadIdx.x >> 5;
  if (lane == 0) lds[wid] = wsum;
  __syncthreads();
  if (threadIdx.x == 0) {
    float s = 0.0f;
#pragma unroll
    for (int i = 0; i < NTHR / 32; ++i) s += lds[i];
    part[blockIdx.x] = s;
  }
}

// ---------------------------------------------------------------------------
// Kernel 2: one full wave (EXEC all ones) folds the per-block partials and
// writes the scalar result.  WMMA again does the 32-lane reduction.
// ---------------------------------------------------------------------------
__global__ __launch_bounds__(32) void rope_final_kernel(
    const float* __restrict__ part, float* __restrict__ out, int nb) {
  float acc = 0.0f;
  for (int i = threadIdx.x; i < nb; i += 32) acc += part[i];
  float total = wave_sum32(acc);
  if (threadIdx.x == 0) out[0] = total;
}

extern "C" void kernel_launch(void* const* d_in, const int* in_sizes, int n_in,
                              void* d_out, int out_size, void* d_ws, size_t ws_size,
                              hipStream_t stream) {
  const float* vec    = (const float*)d_in[0];
  const float* thetas = (const float*)d_in[1];
  const int*   m      = (const int*)d_in[2];
  float*       out    = (float*)d_out;
  float*       part   = (float*)d_ws;   // NBLK floats = 8 KB of scratch
  const int    n      = in_sizes[0];

  rope_partial_kernel<<<NBLK, NTHR, 0, stream>>>(vec, thetas, m, part, n);
  rope_final_kernel<<<1, 32, 0, stream>>>(part, out, NBLK);
}